// path_generator_BS_33603824123992
// MI455X (gfx1250) — compile-verified
//
#include <hip/hip_runtime.h>

#ifndef __has_builtin
#define __has_builtin(x) 0
#endif

#if defined(__HIP_DEVICE_COMPILE__) && \
    __has_builtin(__builtin_amdgcn_tensor_load_to_lds) && \
    __has_builtin(__builtin_amdgcn_s_wait_tensorcnt)
#define USE_TDM 1
#else
#define USE_TDM 0
#endif

namespace {
constexpr int   COLS     = 366;                                   // TIME_STEPS + 1
constexpr int   WAVES    = 8;                                     // 256-thread block, wave32
constexpr int   RPW      = 4;                                     // rows per wave (TDM double-buffer pipeline)
constexpr int   PAD_F    = 388;                                   // LDS floats per row buffer (16B-aligned stride, >= 384)
constexpr int   CHUNKS   = 3;                                     // 3 * 128 = 384 >= 366 columns
constexpr float S0V      = 100.0f;
constexpr float DRIFT_DT = (0.05f - 0.5f * 0.2f * 0.2f) * (1.0f / 365.0f); // (mean - vol^2/2)*dt
constexpr float VOL_SQDT = 0.2f * 0.052342392f;                   // vol * sqrt(dt)
} // namespace

#if USE_TDM
typedef unsigned int u32x4 __attribute__((ext_vector_type(4)));
typedef int          i32x8 __attribute__((ext_vector_type(8)));
typedef int          i32x4 __attribute__((ext_vector_type(4)));

// Issue a TDM load of one 366-float row (global -> LDS). D# layout per CDNA5 ISA §8.
// 6-arg builtin form (clang-23 / therock-10.0): (g0, g1, g2, g3, g4, cpol).
__device__ __forceinline__ void tdm_load_row(unsigned lds_byte, const float* gsrc) {
    const unsigned long long ga = (unsigned long long)gsrc;
    u32x4 g0;
    g0.x = 1u;                                            // count=1 (valid user descriptor)
    g0.y = lds_byte;                                      // lds_addr (workgroup-relative byte offset)
    g0.z = (unsigned)ga;                                  // global_addr[31:0]
    g0.w = ((unsigned)(ga >> 32) & 0x01FFFFFFu)           // global_addr[56:32]
           | 0x80000000u;                                 // type=2 ("image")
    i32x8 g1;
    g1[0] = 0x00020000;                                   // wg_mask=0, data_size=2 (4 bytes)
    g1[1] = (int)((unsigned)COLS << 16);                  // tensor_dim0[15:0] = 366
    g1[2] = 0x00010000;                                   // tensor_dim0 hi=0, tensor_dim1=1
    g1[3] = (int)((unsigned)COLS << 16);                  // tile_dim0 = 366
    g1[4] = 1;                                            // tile_dim1 = 1, tile_dim2 = 0
    g1[5] = COLS;                                         // tensor_dim0_stride = 366
    g1[6] = 0;
    g1[7] = 0;
    i32x4 gz4 = {0, 0, 0, 0};                             // groups 2/3 unused (2D tensor)
    i32x8 gz8 = {0, 0, 0, 0, 0, 0, 0, 0};                 // trailing group operand (unused)
    __builtin_amdgcn_tensor_load_to_lds(g0, g1, gz4, gz4, gz8, 0);
}
#endif

__global__ __launch_bounds__(WAVES * 32)
void gbm_paths_kernel(const float* __restrict__ Z, float* __restrict__ S, int nrows) {
#if USE_TDM
    __shared__ float smem[WAVES * 2 * PAD_F];
#endif
    const int lane    = (int)(threadIdx.x & 31u);
    const int wave    = __builtin_amdgcn_readfirstlane((int)(threadIdx.x >> 5));
    const int rowBase = (int)blockIdx.x * (WAVES * RPW) + wave * RPW;
    if (rowBase >= nrows) return;
    const bool full = (rowBase + RPW) <= nrows;

#if USE_TDM
    float* const b0 = &smem[wave * (2 * PAD_F)];
    float* const b1 = b0 + PAD_F;
    if (full) {
        tdm_load_row((unsigned)(unsigned long long)b0,
                     Z + (unsigned long long)rowBase * COLS);
    }
#endif

#pragma unroll
    for (int i = 0; i < RPW; ++i) {
        const int row = rowBase + i;
        if (!full && row >= nrows) break;

        const float* zg = Z + (unsigned long long)row * COLS;  // global row (fallback path)
        float* const srow = S + (unsigned long long)row * COLS;

#if USE_TDM
        const float* zl = (i & 1) ? b1 : b0;                   // LDS row (fast path)
        if (full) {
            if (i + 1 < RPW) {
                tdm_load_row((unsigned)(unsigned long long)((i & 1) ? b0 : b1),
                             Z + (unsigned long long)(row + 1) * COLS);
                asm volatile("" ::: "memory");
                __builtin_amdgcn_s_wait_tensorcnt(1);          // in-order: row i's TDM is done
            } else {
                __builtin_amdgcn_s_wait_tensorcnt(0);
            }
            asm volatile("" ::: "memory");
        }
#endif

        float carry = 0.0f;
#pragma unroll
        for (int c = 0; c < CHUNKS; ++c) {
            const int t0 = c * 128 + lane * 4;                 // 4 consecutive columns per lane
            float z0, z1, z2, z3;
#if USE_TDM
            if (full) {
                const float4 zv = *(const float4*)(zl + t0);   // 16B-aligned ds_load_b128
                z0 = zv.x; z1 = zv.y; z2 = zv.z; z3 = zv.w;
            } else
#endif
            {
                z0 = (t0 + 0 < COLS) ? zg[t0 + 0] : 0.0f;
                z1 = (t0 + 1 < COLS) ? zg[t0 + 1] : 0.0f;
                z2 = (t0 + 2 < COLS) ? zg[t0 + 2] : 0.0f;
                z3 = (t0 + 3 < COLS) ? zg[t0 + 3] : 0.0f;
            }
            // log-returns; column 0 contributes 0 (S_0 = S0 exactly), OOB columns contribute 0
            const float x0 = (t0 != 0 && t0 < COLS) ? (DRIFT_DT + VOL_SQDT * z0) : 0.0f;
            const float x1 = (t0 + 1 < COLS) ? (DRIFT_DT + VOL_SQDT * z1) : 0.0f;
            const float x2 = (t0 + 2 < COLS) ? (DRIFT_DT + VOL_SQDT * z2) : 0.0f;
            const float x3 = (t0 + 3 < COLS) ? (DRIFT_DT + VOL_SQDT * z3) : 0.0f;

            // serial 4-element prefix in-lane
            const float p0 = x0;
            const float p1 = p0 + x1;
            const float p2 = p1 + x2;
            const float p3 = p2 + x3;

            // wave32 inclusive scan of lane totals (ds_bpermute shuffles)
            float sc = p3;
#pragma unroll
            for (int d = 1; d < 32; d <<= 1) {
                const float y = __shfl_up(sc, (unsigned)d, 32);
                if (lane >= d) sc += y;
            }
            const float base = carry + (sc - p3);              // carry + exclusive prefix

            const float e0 = S0V * __expf(base + p0);
            const float e1 = S0V * __expf(base + p1);
            const float e2 = S0V * __expf(base + p2);
            const float e3 = S0V * __expf(base + p3);

            // 8B-aligned paired stores (row*366 is even, t0 % 4 == 0)
            if (t0 + 1 < COLS) {
                *(float2*)(srow + t0) = make_float2(e0, e1);
            } else if (t0 < COLS) {
                srow[t0] = e0;
            }
            if (t0 + 3 < COLS) {
                *(float2*)(srow + t0 + 2) = make_float2(e2, e3);
            } else if (t0 + 2 < COLS) {
                srow[t0 + 2] = e2;
            }

            carry += __shfl(sc, 31, 32);                       // wave total -> next chunk
        }
    }
}

extern "C" void kernel_launch(void* const* d_in, const int* in_sizes, int n_in,
                              void* d_out, int out_size, void* d_ws, size_t ws_size,
                              hipStream_t stream) {
    const float* Z = (const float*)d_in[0];
    float* S = (float*)d_out;
    const int nrows = in_sizes[0] / COLS;                      // 262144
    const int rows_per_block = WAVES * RPW;                    // 32
    const int grid = (nrows + rows_per_block - 1) / rows_per_block;
    gbm_paths_kernel<<<dim3(grid), dim3(WAVES * 32), 0, stream>>>(Z, S, nrows);
}